// CenterCut2_42984032698899
// MI455X (gfx1250) — compile-verified
//
#include <hip/hip_runtime.h>
#include <stdint.h>

// ---------------------------------------------------------------------------
// CenterCut2: per-sample power argmax -> circular roll to center -> crop.
// roll+crop == wrap-around gather centered at the peak, so we never roll.
//
// Phase 1 (bandwidth bound, 256 MB stream): per-sample argmax of x0^2+x1^2
//   using the gfx1250 async global->LDS copy path (4-deep per-wave ring,
//   s_wait_asynccnt pipelining), packed (power_bits<<32 | ~idx) keys,
//   wave32 shfl reduction + LDS + global_atomic_max_u64.
// Phase 2 (8 MB): direct wrap-around gather of the 16x32x32 window.
// ---------------------------------------------------------------------------

typedef float v4f __attribute__((ext_vector_type(4)));
typedef int   v4i __attribute__((vector_size(16)));   // matches builtin param

#define B_    32
#define C_    2
#define D_    64
#define H_    128
#define W_    128
#define NVOX  (D_ * H_ * W_)          // 1048576 voxels per sample
#define RD    8
#define RH    16
#define RW    16
#define OD    (2 * RD)                // 16
#define OH    (2 * RH)                // 32
#define OW    (2 * RW)                // 32

#define P1_WGS  32                    // workgroups per sample
#define P1_T    256                   // threads per workgroup (8 waves)
#define CHUNK4  ((NVOX / 4) / P1_WGS) // 8192 float4 per workgroup
#define ITERS   (CHUNK4 / P1_T)       // 32 iterations per thread (exact)
#define PIPE    4                     // async pipeline depth (ring buffers)

#if defined(__HIP_DEVICE_COMPILE__) && \
    __has_builtin(__builtin_amdgcn_global_load_async_to_lds_b128) && \
    __has_builtin(__builtin_amdgcn_s_wait_asynccnt)
#define USE_ASYNC_LDS 1
#else
#define USE_ASYNC_LDS 0
#endif

#if USE_ASYNC_LDS
// Issue one 16B async global->LDS copy for this lane.
// Global ptr: flat and AS1 addresses are numerically identical for global
// memory, so reinterpret via uintptr_t. LDS ptr: real addrspacecast (flat LDS
// addresses carry the aperture base, AS3 addresses are offsets).
__device__ __forceinline__ void async_copy_b128(const v4f* gsrc, v4f* ldst) {
    __attribute__((address_space(1))) v4i* g =
        (__attribute__((address_space(1))) v4i*)(uintptr_t)gsrc;
    v4i* lgen = (v4i*)ldst;
    __attribute__((address_space(3))) v4i* l =
        (__attribute__((address_space(3))) v4i*)lgen;
    __builtin_amdgcn_global_load_async_to_lds_b128(g, l, 0, 0);
}
#endif

__device__ __forceinline__ unsigned long long shfl_xor_u64(unsigned long long v, int m) {
    unsigned int lo = (unsigned int)(v & 0xFFFFFFFFull);
    unsigned int hi = (unsigned int)(v >> 32);
    lo = __shfl_xor(lo, m, 32);
    hi = __shfl_xor(hi, m, 32);
    return (((unsigned long long)hi) << 32) | (unsigned long long)lo;
}

__global__ void cc2_p0_init(unsigned long long* __restrict__ slots) {
    if (threadIdx.x < B_) slots[threadIdx.x] = 0ull;
}

__global__ __launch_bounds__(P1_T) void cc2_p1_argmax(
    const float* __restrict__ x, unsigned long long* __restrict__ slots) {
    const int b    = blockIdx.y;
    const int tid  = threadIdx.x;
    const int lane = tid & 31;
    const int wave = tid >> 5;

    const v4f* __restrict__ ch0 = (const v4f*)(x + (size_t)b * (size_t)(C_ * NVOX));
    const v4f* __restrict__ ch1 = ch0 + (NVOX / 4);

    const unsigned int base4 = blockIdx.x * CHUNK4 + (unsigned int)tid;

    float        bestP = -1.0f;   // every thread always sees data; power >= 0
    unsigned int bestI = 0u;

#if USE_ASYNC_LDS
    // Per-wave ring staging: [wave][buf][chan][lane], 8*4*2*32*16B = 32 KB.
    __shared__ v4f stage[P1_T / 32][PIPE][2][32];

    // Prologue: stage iterations 0..PIPE-2 (2*(PIPE-1) = 6 loads in flight).
    #pragma unroll
    for (int j = 0; j < PIPE - 1; ++j) {
        const unsigned int i4 = base4 + (unsigned int)j * P1_T;
        async_copy_b128(ch0 + i4, &stage[wave][j][0][lane]);
        async_copy_b128(ch1 + i4, &stage[wave][j][1][lane]);
    }

    for (int k = 0; k < ITERS; ++k) {
        if (k + PIPE - 1 < ITERS) {
            const unsigned int i4n = base4 + (unsigned int)(k + PIPE - 1) * P1_T;
            const int buf = (k + PIPE - 1) & (PIPE - 1);
            async_copy_b128(ch0 + i4n, &stage[wave][buf][0][lane]);
            async_copy_b128(ch1 + i4n, &stage[wave][buf][1][lane]);
            // loads retire in order: <= 2*(PIPE-1) outstanding => iter k done
            __builtin_amdgcn_s_wait_asynccnt(2 * (PIPE - 1));
        } else {
            // drain: immediates must be compile-time constants
            const int rem = ITERS - 1 - k;      // PIPE-2 .. 0
            if      (rem >= 3) __builtin_amdgcn_s_wait_asynccnt(6);
            else if (rem == 2) __builtin_amdgcn_s_wait_asynccnt(4);
            else if (rem == 1) __builtin_amdgcn_s_wait_asynccnt(2);
            else               __builtin_amdgcn_s_wait_asynccnt(0);
        }
        const v4f a = stage[wave][k & (PIPE - 1)][0][lane];
        const v4f c = stage[wave][k & (PIPE - 1)][1][lane];
        const unsigned int i0 = (base4 + (unsigned int)k * P1_T) * 4u;
        const float p0 = a.x * a.x + c.x * c.x;
        const float p1 = a.y * a.y + c.y * c.y;
        const float p2 = a.z * a.z + c.z * c.z;
        const float p3 = a.w * a.w + c.w * c.w;
        if (p0 > bestP) { bestP = p0; bestI = i0 + 0u; }
        if (p1 > bestP) { bestP = p1; bestI = i0 + 1u; }
        if (p2 > bestP) { bestP = p2; bestI = i0 + 2u; }
        if (p3 > bestP) { bestP = p3; bestI = i0 + 3u; }
    }
#else
    #pragma unroll 4
    for (int k = 0; k < ITERS; ++k) {
        const unsigned int i4 = base4 + (unsigned int)k * P1_T;
        const v4f a = __builtin_nontemporal_load(ch0 + i4);
        const v4f c = __builtin_nontemporal_load(ch1 + i4);
        const unsigned int i0 = i4 * 4u;
        const float p0 = a.x * a.x + c.x * c.x;
        const float p1 = a.y * a.y + c.y * c.y;
        const float p2 = a.z * a.z + c.z * c.z;
        const float p3 = a.w * a.w + c.w * c.w;
        if (p0 > bestP) { bestP = p0; bestI = i0 + 0u; }
        if (p1 > bestP) { bestP = p1; bestI = i0 + 1u; }
        if (p2 > bestP) { bestP = p2; bestI = i0 + 2u; }
        if (p3 > bestP) { bestP = p3; bestI = i0 + 3u; }
    }
#endif

    // Pack: power bits (monotone u32 since power >= 0) | ~idx (ties -> min idx).
    unsigned long long key =
        (((unsigned long long)__float_as_uint(bestP)) << 32) |
        (unsigned long long)(unsigned int)(~bestI);

    // wave32 max reduction
    for (int m = 16; m > 0; m >>= 1) {
        const unsigned long long o = shfl_xor_u64(key, m);
        key = (o > key) ? o : key;
    }

    __shared__ unsigned long long wred[P1_T / 32];
    if (lane == 0) wred[wave] = key;
    __syncthreads();
    if (tid == 0) {
        unsigned long long k = wred[0];
        #pragma unroll
        for (int i = 1; i < P1_T / 32; ++i) k = (wred[i] > k) ? wred[i] : k;
        atomicMax(&slots[b], k);   // global_atomic_max_u64 across workgroups
    }
}

__global__ __launch_bounds__(256) void cc2_p2_gather(
    const float* __restrict__ x, const unsigned long long* __restrict__ slots,
    float* __restrict__ out, int n) {
    const unsigned int t = blockIdx.x * 256u + threadIdx.x;
    if (t >= (unsigned int)n) return;

    // out layout (B, C, OD, OH, OW); all dims pow2 -> bitfield decode
    const unsigned int ow = t & (OW - 1);
    const unsigned int oh = (t >> 5) & (OH - 1);
    const unsigned int od = (t >> 10) & (OD - 1);
    const unsigned int c  = (t >> 14) & (C_ - 1);
    const unsigned int b  = t >> 15;

    const unsigned long long key = slots[b];
    const unsigned int idx = ~((unsigned int)(key & 0xFFFFFFFFull));
    const unsigned int d = idx >> 14;          // / (H*W) = / 16384
    const unsigned int h = (idx >> 7) & (H_ - 1);
    const unsigned int w = idx & (W_ - 1);

    // rolled[center-r+o] == x[(peak - r + o) mod N]
    const unsigned int sd = (d + (D_ - RD) + od) & (D_ - 1);
    const unsigned int sh = (h + (H_ - RH) + oh) & (H_ - 1);
    const unsigned int sw = (w + (W_ - RW) + ow) & (W_ - 1);

    const size_t src = ((((size_t)b * C_ + c) * D_ + sd) * H_ + sh) * W_ + sw;
    out[t] = x[src];
}

extern "C" void kernel_launch(void* const* d_in, const int* in_sizes, int n_in,
                              void* d_out, int out_size, void* d_ws, size_t ws_size,
                              hipStream_t stream) {
    const float* x = (const float*)d_in[0];
    float* out = (float*)d_out;
    unsigned long long* slots = (unsigned long long*)d_ws;  // 32 * 8 bytes

    cc2_p0_init<<<1, 32, 0, stream>>>(slots);
    cc2_p1_argmax<<<dim3(P1_WGS, B_), P1_T, 0, stream>>>(x, slots);
    cc2_p2_gather<<<(out_size + 255) / 256, 256, 0, stream>>>(x, slots, out, out_size);
}